// Attention_42279658062045
// MI455X (gfx1250) — compile-verified
//
#include <hip/hip_runtime.h>
#include <cmath>

typedef __attribute__((ext_vector_type(2))) float v2f;
typedef __attribute__((ext_vector_type(8))) float v8f;

#define KN 30          // neighbors
#define KP 32          // padded K
#define DN 64          // feature dim
#define ROWS_PER_WAVE 16
#define WAVES_PER_BLOCK 4
#define BLOCK_THREADS (WAVES_PER_BLOCK * 32)

__global__ __launch_bounds__(BLOCK_THREADS)
void attn_fused_kernel(const float* __restrict__ sd,     // [B, KN]
                       const float* __restrict__ ctx,    // [B, KN, DN]
                       const float* __restrict__ kern,   // [KN, KN]
                       const float* __restrict__ bias,   // [KN]
                       float* __restrict__ out,          // [B, DN]
                       int B)
{
    __shared__ float kern_lds[KP][KP];                                  // zero-padded 30x30
    __shared__ float simi_lds[WAVES_PER_BLOCK][ROWS_PER_WAVE][KP];      // zero-padded
    __shared__ float w_lds[WAVES_PER_BLOCK][ROWS_PER_WAVE][KP];         // softmax weights

    const int tid  = threadIdx.x;
    const int wave = tid >> 5;
    const int lane = tid & 31;

    // ---- Block cooperative: zero-padded kernel matrix into LDS (3.6KB live) ----
    for (int i = tid; i < KP * KP; i += BLOCK_THREADS) {
        int r = i >> 5, c = i & 31;
        kern_lds[r][c] = (r < KN && c < KN) ? kern[r * KN + c] : 0.0f;
    }

    const int b0 = (blockIdx.x * WAVES_PER_BLOCK + wave) * ROWS_PER_WAVE;

    // ---- Per wave: Gaussian compatibility simi = exp(-d^2/2), padded K -> 0 ----
    for (int i = lane; i < ROWS_PER_WAVE * KP; i += 32) {
        int m = i >> 5, k = i & 31;
        int b = b0 + m; if (b >= B) b = B - 1;          // clamp keeps EXEC full
        float v = 0.0f;
        if (k < KN) { float d = sd[b * KN + k]; v = __expf(-0.5f * d * d); }
        simi_lds[wave][m][k] = v;
    }
    __syncthreads();

    // ---- WMMA: logits[16 rows x 32 cols] = simi(16x32) @ kernel(32x32) + bias ----
    // f32 16x16x4 operand layout: lane L -> M = L&15, K slot = koff(+1), koff = 2*(L>=16)
    const int mrow = lane & 15;
    const int koff = (lane >> 4) << 1;
    const int n0   = lane & 15;          // N tile 0: columns 0..15
    const int n1   = n0 + 16;            // N tile 1: columns 16..31 (>=30 masked later)

    v8f acc0, acc1;
    {
        float bz0 = bias[n0];
        float bz1 = (n1 < KN) ? bias[n1] : 0.0f;
        #pragma unroll
        for (int r = 0; r < 8; ++r) { acc0[r] = bz0; acc1[r] = bz1; }
    }

    #pragma unroll
    for (int c = 0; c < 8; ++c) {        // K = 32 in chunks of 4
        const int kb = 4 * c + koff;
        v2f a;  a.x  = simi_lds[wave][mrow][kb]; a.y  = simi_lds[wave][mrow][kb + 1];
        v2f bA; bA.x = kern_lds[kb][n0];         bA.y = kern_lds[kb + 1][n0];
        v2f bB; bB.x = kern_lds[kb][n1];         bB.y = kern_lds[kb + 1][n1];
        acc0 = __builtin_amdgcn_wmma_f32_16x16x4_f32(false, a, false, bA, (short)0, acc0, false, false);
        acc1 = __builtin_amdgcn_wmma_f32_16x16x4_f32(false, a, false, bB, (short)0, acc1, false, false);
    }

    // ---- Softmax over 30 columns, per row, in-register (width-16 shuffles) ----
    // C/D layout: VGPR r, lanes 0-15 -> row r, lanes 16-31 -> row r+8, N = lane&15 (+16 for tile1)
    const bool valid1 = (n1 < KN);
    #pragma unroll
    for (int r = 0; r < 8; ++r) {
        float v0 = acc0[r];
        float v1 = valid1 ? acc1[r] : -__builtin_inff();
        float mx = fmaxf(v0, v1);
        #pragma unroll
        for (int off = 8; off >= 1; off >>= 1)
            mx = fmaxf(mx, __shfl_xor(mx, off, 16));
        float e0 = __expf(v0 - mx);
        float e1 = valid1 ? __expf(acc1[r] - mx) : 0.0f;
        float s = e0 + e1;
        #pragma unroll
        for (int off = 8; off >= 1; off >>= 1)
            s += __shfl_xor(s, off, 16);
        float inv = 1.0f / s;
        int m = (lane < 16) ? r : (r + 8);
        w_lds[wave][m][n0] = e0 * inv;
        w_lds[wave][m][n1] = e1 * inv;   // zero for n1 >= 30
    }
    __syncthreads();

    // ---- Bandwidth phase: mean[b,:] = sum_k w[b,k] * context[b,k,:] ----
    // lane covers d = 2*lane, 2*lane+1 -> one fully coalesced 256B slab per (b,k)
    for (int m = 0; m < ROWS_PER_WAVE; ++m) {
        int b = b0 + m;
        if (b >= B) break;
        const float* base = ctx + ((size_t)b * KN) * DN + 2 * lane;
        float ax = 0.0f, ay = 0.0f;
        #pragma unroll
        for (int k = 0; k < KN; ++k) {
            float  w = w_lds[wave][m][k];           // LDS broadcast read
            float2 c = *(const float2*)(base + k * DN);
            ax = fmaf(w, c.x, ax);
            ay = fmaf(w, c.y, ay);
        }
        float2 o; o.x = ax; o.y = ay;
        *(float2*)(out + (size_t)b * DN + 2 * lane) = o;
    }
}

extern "C" void kernel_launch(void* const* d_in, const int* in_sizes, int n_in,
                              void* d_out, int out_size, void* d_ws, size_t ws_size,
                              hipStream_t stream) {
    const float* sd   = (const float*)d_in[0];   // [B, 30]
    const float* ctx  = (const float*)d_in[1];   // [B, 30, 64]
    const float* kern = (const float*)d_in[2];   // [30, 30]
    const float* bias = (const float*)d_in[3];   // [30]
    float* out = (float*)d_out;                  // [B, 64]

    const int B = in_sizes[0] / KN;
    const int rows_per_block = WAVES_PER_BLOCK * ROWS_PER_WAVE;   // 64
    const int grid = (B + rows_per_block - 1) / rows_per_block;   // 3125 for B=200000

    attn_fused_kernel<<<grid, BLOCK_THREADS, 0, stream>>>(sd, ctx, kern, bias, out, B);
}